// MultiHeadAttentionModule_21165598835360
// MI455X (gfx1250) — compile-verified
//
#include <hip/hip_runtime.h>

typedef __attribute__((ext_vector_type(2))) float v2f;
typedef __attribute__((ext_vector_type(4))) float v4f;
typedef __attribute__((ext_vector_type(8))) float v8f;

#define D_MODEL 1024
#define H_HEADS 16
#define HEAD_DIM 64
#define SEQ_L 2048
#define BATCH 2
#define M_ROWS (BATCH * SEQ_L) /* 4096 */

// ---------------------------------------------------------------------------
// fp32 WMMA: D(16x16) = A(16x4) * B(4x16) + C, full fp32 precision.
// A frag: lane holds A[lane&15][2*(lane>>4) + {0,1}]
// B frag: lane holds B[2*(lane>>4) + {0,1}][lane&15]
// C/D:    vgpr v, lane l -> C[v + 8*(l>>4)][l&15]
// ---------------------------------------------------------------------------
__device__ __forceinline__ v8f wmma_f32(v2f a, v2f b, v8f c) {
  return __builtin_amdgcn_wmma_f32_16x16x4_f32(false, a, false, b, (short)0, c,
                                               false, false);
}

__device__ __forceinline__ float rowmax16(float x) {
#pragma unroll
  for (int m = 1; m < 16; m <<= 1) x = fmaxf(x, __shfl_xor(x, m, 32));
  return x;
}
__device__ __forceinline__ float rowsum16(float x) {
#pragma unroll
  for (int m = 1; m < 16; m <<= 1) x += __shfl_xor(x, m, 32);
  return x;
}

// ---------------------------------------------------------------------------
// GEMM:  C[M,N] = A[M,K] @ B[K,N] + bias[N]    (row-major, fp32)
// Block tile 64x128, 8 waves (2x4), wave tile 32x32 (2x2 WMMA accumulators).
// ---------------------------------------------------------------------------
#define BM 64
#define BN 128
#define BK 32
#define APAD 36 /* 144B row stride: b128-store aligned, conflict-free b64 reads */
#define BPAD 34 /* 136B row stride: b64-read aligned */

__global__ __launch_bounds__(256) void gemm_bias_kernel(
    const float* __restrict__ A, const float* __restrict__ B,
    const float* __restrict__ bias, float* __restrict__ C, int N, int K) {
  __shared__ float As[BM][APAD];
  __shared__ float Bs[BN][BPAD]; // stored transposed: Bs[n][k]

  const int tid = threadIdx.x;
  const int lane = tid & 31;
  const int w = tid >> 5;
  const int wm = (w >> 2) << 5; // 0 / 32
  const int wn = (w & 3) << 5;  // 0..96
  const int m0 = blockIdx.y * BM;
  const int n0 = blockIdx.x * BN;

  const int ln = lane & 15;
  const int half = lane >> 4;
  const int kb = half << 1;

  v8f acc[2][2];
#pragma unroll
  for (int i = 0; i < 2; ++i)
#pragma unroll
    for (int j = 0; j < 2; ++j)
#pragma unroll
      for (int v = 0; v < 8; ++v) acc[i][j][v] = 0.0f;

  for (int k0 = 0; k0 < K; k0 += BK) {
    __syncthreads();
    { // A tile 64x32, b128 stores
      int r = tid >> 3;
      int c = (tid & 7) << 2;
      const float* ap = A + (size_t)(m0 + r) * K + (k0 + c);
      *(v4f*)&As[r][c] = *(const v4f*)ap;
      *(v4f*)&As[r + 32][c] = *(const v4f*)(ap + (size_t)32 * K);
    }
    { // B tile 32x128, coalesced global reads, transposed LDS stores
      int kr = tid >> 5;
      int nc = (tid & 31) << 2;
#pragma unroll
      for (int it = 0; it < 4; ++it) {
        int kk = kr + it * 8;
        v4f bv = *(const v4f*)(B + (size_t)(k0 + kk) * N + (n0 + nc));
        Bs[nc + 0][kk] = bv.x;
        Bs[nc + 1][kk] = bv.y;
        Bs[nc + 2][kk] = bv.z;
        Bs[nc + 3][kk] = bv.w;
      }
    }
    __syncthreads();
#pragma unroll
    for (int kk = 0; kk < BK; kk += 4) {
      v2f a0 = *(const v2f*)&As[wm + ln][kk + kb];
      v2f a1 = *(const v2f*)&As[wm + 16 + ln][kk + kb];
      v2f b0 = *(const v2f*)&Bs[wn + ln][kk + kb];
      v2f b1 = *(const v2f*)&Bs[wn + 16 + ln][kk + kb];
      acc[0][0] = wmma_f32(a0, b0, acc[0][0]);
      acc[0][1] = wmma_f32(a0, b1, acc[0][1]);
      acc[1][0] = wmma_f32(a1, b0, acc[1][0]);
      acc[1][1] = wmma_f32(a1, b1, acc[1][1]);
    }
  }

#pragma unroll
  for (int j = 0; j < 2; ++j) {
    float bj = bias[n0 + wn + j * 16 + ln];
#pragma unroll
    for (int i = 0; i < 2; ++i) {
      int rbase = m0 + wm + i * 16 + half * 8;
      int col = n0 + wn + j * 16 + ln;
#pragma unroll
      for (int v = 0; v < 8; ++v)
        C[(size_t)(rbase + v) * N + col] = acc[i][j][v] + bj;
    }
  }
}

// ---------------------------------------------------------------------------
// Causal flash attention. One block = (batch,head, 64-query block); 4 waves,
// each wave owns a 16-row Q tile kept in registers as WMMA A-fragments.
// qkv layout: [b, l, 3*1024] with Q at +0, K at +1024, V at +2048, per-head
// slices of 64. Output written as [b, l, h*64+d] (== [4096,1024]).
// ---------------------------------------------------------------------------
#define KPAD 68 /* 272B row stride: b128-store aligned, conflict-free b64 reads */

__global__ __launch_bounds__(128) void mha_kernel(const float* __restrict__ qkv,
                                                  float* __restrict__ attn_out) {
  __shared__ float Ks[64][KPAD];     // K tile, row-major [kv][d]
  __shared__ float Vs[64][KPAD];     // V tile, transposed [d][kv]
  __shared__ float Ps[4][16][KPAD];  // per-wave P tile, [qrow][kv]

  const int qb = blockIdx.x;  // query block 0..31
  const int bh = blockIdx.y;  // b*16 + h
  const int b = bh >> 4;
  const int h = bh & 15;

  const int tid = threadIdx.x;
  const int lane = tid & 31;
  const int w = tid >> 5; // 0..3
  const int ln = lane & 15;
  const int half = lane >> 4;
  const int kb = half << 1;

  const int qw0 = qb * 64 + w * 16;
  const size_t rstride = 3 * D_MODEL;
  const float* base = qkv + (size_t)b * SEQ_L * rstride + (size_t)h * HEAD_DIM;

  // Q fragments, resident for the whole KV loop
  v2f qa[16];
  {
    const float* qp = base + (size_t)(qw0 + ln) * rstride + kb;
#pragma unroll
    for (int kk = 0; kk < 16; ++kk) qa[kk] = *(const v2f*)(qp + kk * 4);
  }

  v8f acc_o[4];
  float m_i[8], l_i[8];
#pragma unroll
  for (int v = 0; v < 8; ++v) {
    m_i[v] = -__builtin_inff();
    l_i[v] = 0.0f;
    acc_o[0][v] = 0.0f; acc_o[1][v] = 0.0f;
    acc_o[2][v] = 0.0f; acc_o[3][v] = 0.0f;
  }

  const float scale = 0.03125f; // 1/sqrt(1024) (reference scales by d_model)
  const float log2e = 1.44269504088896f;

  for (int j = 0; j <= qb; ++j) {
    __syncthreads();
    { // stage K (row-major) and V (transposed) tiles
      int r0 = tid >> 4;
      int c = (tid & 15) << 2;
#pragma unroll
      for (int it = 0; it < 8; ++it) {
        int r = r0 + it * 8;
        const float* row = base + (size_t)(j * 64 + r) * rstride;
        *(v4f*)&Ks[r][c] = *(const v4f*)(row + D_MODEL + c);
        v4f vv = *(const v4f*)(row + 2 * D_MODEL + c);
        Vs[c + 0][r] = vv.x;
        Vs[c + 1][r] = vv.y;
        Vs[c + 2][r] = vv.z;
        Vs[c + 3][r] = vv.w;
      }
    }
    __syncthreads();

    // S = Q @ K^T  (16x64 per wave, 4 accumulators)
    v8f s[4];
#pragma unroll
    for (int nt = 0; nt < 4; ++nt)
#pragma unroll
      for (int v = 0; v < 8; ++v) s[nt][v] = 0.0f;
#pragma unroll
    for (int kk = 0; kk < 16; ++kk) {
      v2f b0 = *(const v2f*)&Ks[0 + ln][kk * 4 + kb];
      v2f b1 = *(const v2f*)&Ks[16 + ln][kk * 4 + kb];
      v2f b2 = *(const v2f*)&Ks[32 + ln][kk * 4 + kb];
      v2f b3 = *(const v2f*)&Ks[48 + ln][kk * 4 + kb];
      s[0] = wmma_f32(qa[kk], b0, s[0]);
      s[1] = wmma_f32(qa[kk], b1, s[1]);
      s[2] = wmma_f32(qa[kk], b2, s[2]);
      s[3] = wmma_f32(qa[kk], b3, s[3]);
    }

    const bool do_mask = (j == qb); // only diagonal tile crosses the causal edge
    float tmax[8];
#pragma unroll
    for (int v = 0; v < 8; ++v) {
      float mx = -__builtin_inff();
#pragma unroll
      for (int nt = 0; nt < 4; ++nt) {
        float sv = s[nt][v] * scale;
        if (do_mask) {
          int kvg = j * 64 + nt * 16 + ln;
          int qg = qw0 + v + half * 8;
          if (kvg > qg) sv = -__builtin_inff();
        }
        s[nt][v] = sv;
        mx = fmaxf(mx, sv);
      }
      tmax[v] = rowmax16(mx);
    }

#pragma unroll
    for (int v = 0; v < 8; ++v) {
      float mn = fmaxf(m_i[v], tmax[v]);
      float alpha = exp2f((m_i[v] - mn) * log2e);
      m_i[v] = mn;
      float rs = 0.0f;
#pragma unroll
      for (int nt = 0; nt < 4; ++nt) {
        float p = exp2f((s[nt][v] - mn) * log2e);
        rs += p;
        Ps[w][v + half * 8][nt * 16 + ln] = p; // D-layout -> LDS
      }
      l_i[v] = l_i[v] * alpha + rowsum16(rs);
      acc_o[0][v] *= alpha;
      acc_o[1][v] *= alpha;
      acc_o[2][v] *= alpha;
      acc_o[3][v] *= alpha;
    }

    // wave-private LDS round trip: wait our own ds stores before A-frag reads
    asm volatile("s_wait_dscnt 0" ::: "memory");

    // O += P @ V
#pragma unroll
    for (int kk = 0; kk < 16; ++kk) {
      v2f pa = *(const v2f*)&Ps[w][ln][kk * 4 + kb];
      v2f b0 = *(const v2f*)&Vs[0 + ln][kk * 4 + kb];
      v2f b1 = *(const v2f*)&Vs[16 + ln][kk * 4 + kb];
      v2f b2 = *(const v2f*)&Vs[32 + ln][kk * 4 + kb];
      v2f b3 = *(const v2f*)&Vs[48 + ln][kk * 4 + kb];
      acc_o[0] = wmma_f32(pa, b0, acc_o[0]);
      acc_o[1] = wmma_f32(pa, b1, acc_o[1]);
      acc_o[2] = wmma_f32(pa, b2, acc_o[2]);
      acc_o[3] = wmma_f32(pa, b3, acc_o[3]);
    }
  }

  // normalize and write [b, q, h*64 + d]
  float* obase = attn_out + (size_t)b * SEQ_L * D_MODEL + (size_t)h * HEAD_DIM;
#pragma unroll
  for (int v = 0; v < 8; ++v) {
    int qg = qw0 + v + half * 8;
    float inv_l = 1.0f / l_i[v];
#pragma unroll
    for (int nt = 0; nt < 4; ++nt)
      obase[(size_t)qg * D_MODEL + nt * 16 + ln] = acc_o[nt][v] * inv_l;
  }
}

// ---------------------------------------------------------------------------
extern "C" void kernel_launch(void* const* d_in, const int* in_sizes, int n_in,
                              void* d_out, int out_size, void* d_ws,
                              size_t ws_size, hipStream_t stream) {
  (void)in_sizes; (void)n_in; (void)out_size; (void)ws_size;
  const float* x = (const float*)d_in[0];
  const float* w_qkv = (const float*)d_in[1];
  const float* b_qkv = (const float*)d_in[2];
  const float* w_out = (const float*)d_in[3];
  const float* b_out = (const float*)d_in[4];
  float* out = (float*)d_out;

  float* qkv = (float*)d_ws;                            // [4096, 3072]
  float* attn = qkv + (size_t)M_ROWS * 3 * D_MODEL;     // [4096, 1024]

  // 1) QKV projection: [4096,1024] @ [1024,3072] + b_qkv
  gemm_bias_kernel<<<dim3((3 * D_MODEL) / BN, M_ROWS / BM), 256, 0, stream>>>(
      x, w_qkv, b_qkv, qkv, 3 * D_MODEL, D_MODEL);

  // 2) causal MHA: grid (query blocks, batch*heads)
  mha_kernel<<<dim3(SEQ_L / 64, BATCH * H_HEADS), 128, 0, stream>>>(qkv, attn);

  // 3) output projection: [4096,1024] @ [1024,1024] + b_out
  gemm_bias_kernel<<<dim3(D_MODEL / BN, M_ROWS / BM), 256, 0, stream>>>(
      attn, w_out, b_out, out, D_MODEL, D_MODEL);
}